// LatentTree_25271587570082
// MI455X (gfx1250) — compile-verified
//
#include <hip/hip_runtime.h>

// ---------------------------------------------------------------------------
// LatentTree on gfx1250:
//   XA = x @ A.T  (fp32 WMMA 16x16x4),  then per-row BST path-min + clamp.
// x: [N,512] f32, A: [63,512] f32, out: [N,127] f32
//
// Wave tile: 32 rows x 64 cols (2 A-fragments x 4 B-fragments -> 8 WMMA/k-step)
// Block: 8 waves = 256 rows. A staged in LDS in K-chunks of 128.
// ---------------------------------------------------------------------------

typedef float v2f __attribute__((ext_vector_type(2)));
typedef float v8f __attribute__((ext_vector_type(8)));

#define KDIM        512
#define KCHUNK      128     // K-slice of A staged in LDS
#define NSPLIT      63
#define NODES       127
#define ROWS_WAVE   32
#define WAVES       8
#define ROWS_BLOCK  (ROWS_WAVE * WAVES)   // 256
#define ASTRIDE     132     // LDS float stride for A chunk rows (bank-conflict free)
#define XSTRIDE     66      // LDS float stride for per-wave XA tile rows (16 rows/pass)
// LDS: max(64*132, 8*16*66) = 8448 floats = 33792 bytes (both uses identical)
#define LDS_FLOATS  (64 * ASTRIDE)

__global__ __launch_bounds__(256) void latent_tree_kernel(
    const float* __restrict__ x,
    const float* __restrict__ A,
    float* __restrict__ out)
{
    __shared__ float lds[LDS_FLOATS];

    const int tid  = threadIdx.x;
    const int wave = tid >> 5;
    const int lane = tid & 31;
    const int half = lane >> 4;   // 0: lanes 0-15, 1: lanes 16-31
    const int l16  = lane & 15;

    const int rowbase = blockIdx.x * ROWS_BLOCK + wave * ROWS_WAVE;

    // Per-lane base pointers for A-fragment loads of x (16x4 f32 layout):
    // lane group 0 holds K = k0,k0+1 ; lane group 1 holds K = k0+2,k0+3
    const float* xpLo = x + (size_t)(rowbase + l16) * KDIM + 2 * half;
    const float* xpHi = xpLo + (size_t)16 * KDIM;

    v8f acc[2][4];
    #pragma unroll
    for (int g = 0; g < 2; ++g)
        #pragma unroll
        for (int c = 0; c < 4; ++c)
            acc[g][c] = (v8f){};

    for (int kc = 0; kc < KDIM; kc += KCHUNK) {
        __syncthreads();  // previous chunk readers done before overwrite

        // ---- stage A[:, kc:kc+128] into LDS (row 63 zero-padded), coalesced
        // 64 rows * 32 float4 = 2048 float4 ops / 256 threads = 8 each
        #pragma unroll
        for (int i = 0; i < 8; ++i) {
            const int idx = tid + i * 256;   // float4 index within chunk
            const int n   = idx >> 5;        // row 0..63
            const int j   = idx & 31;        // float4 column 0..31
            float4 v = make_float4(0.f, 0.f, 0.f, 0.f);
            if (n < NSPLIT)
                v = *(const float4*)(A + (size_t)n * KDIM + kc + 4 * j);
            *(float4*)(&lds[n * ASTRIDE + 4 * j]) = v;
        }
        __syncthreads();

        // ---- K inner loop: 32 steps of K=4, 8 WMMA per step
        #pragma unroll 4
        for (int ks = 0; ks < KCHUNK; ks += 4) {
            // A-fragments (x rows), 2 consecutive f32 per lane, 2 row blocks
            v2f aLo = *(const v2f*)(xpLo + kc + ks);
            v2f aHi = *(const v2f*)(xpHi + kc + ks);

            // B-fragments (A^T cols): lane n holds column N=n, K pair per half
            const float* bb = &lds[ks + 2 * half];
            v2f b[4];
            #pragma unroll
            for (int c = 0; c < 4; ++c)
                b[c] = *(const v2f*)(bb + (l16 + 16 * c) * ASTRIDE);

            #pragma unroll
            for (int c = 0; c < 4; ++c) {
                acc[0][c] = __builtin_amdgcn_wmma_f32_16x16x4_f32(
                    false, aLo, false, b[c], (short)0, acc[0][c], false, false);
                acc[1][c] = __builtin_amdgcn_wmma_f32_16x16x4_f32(
                    false, aHi, false, b[c], (short)0, acc[1][c], false, false);
            }
        }
    }

    __syncthreads();  // all waves done with A region; repurpose LDS

    // ---- epilogue in two half-passes of 16 rows through per-wave LDS scratch
    float* xs = &lds[wave * (16 * XSTRIDE)];

    #pragma unroll
    for (int g = 0; g < 2; ++g) {
        // spill XA half-tile: xs[16 rows][64 cols] (stride 66)
        #pragma unroll
        for (int r = 0; r < 8; ++r) {
            const int m = r + 8 * half;   // C layout: vgpr r, lanes 16-31 -> M=r+8
            xs[m * XSTRIDE +  0 + l16] = acc[g][0][r];
            xs[m * XSTRIDE + 16 + l16] = acc[g][1][r];
            xs[m * XSTRIDE + 32 + l16] = acc[g][2][r];
            xs[m * XSTRIDE + 48 + l16] = acc[g][3][r];
        }
        // same-wave DS ops are in order; backend inserts the dscnt wait.

        // per-row BST path-min + clamp, lane-per-node
        for (int r = 0; r < 16; ++r) {
            const float* srow = &xs[r * XSTRIDE];
            const size_t orow = (size_t)(rowbase + g * 16 + r) * NODES;
            #pragma unroll
            for (int c = 0; c < 4; ++c) {
                const int v = c * 32 + lane;
                if (v < NODES) {
                    float q = 1.0f;
                    int u = v;
                    while (u > 0) {
                        const int p = (u - 1) >> 1;
                        const float sp = srow[p];           // parent split score
                        q = fminf(q, (u & 1) ? sp : -sp);   // odd = left child
                        u = p;
                    }
                    out[orow + v] = fminf(fmaxf(q, 0.0f), 1.0f);
                }
            }
        }
    }
}

extern "C" void kernel_launch(void* const* d_in, const int* in_sizes, int n_in,
                              void* d_out, int out_size, void* d_ws, size_t ws_size,
                              hipStream_t stream)
{
    const float* x = (const float*)d_in[0];
    const float* A = (const float*)d_in[1];
    float* out = (float*)d_out;

    const int nrows   = in_sizes[0] / KDIM;         // 131072
    const int nblocks = nrows / ROWS_BLOCK;         // 512

    latent_tree_kernel<<<dim3(nblocks), dim3(256), 0, stream>>>(x, A, out);
}